// NoisyTopkRouter_28870770164343
// MI455X (gfx1250) — compile-verified
//
#include <hip/hip_runtime.h>
#include <hip/hip_bf16.h>
#include <math.h>

typedef __attribute__((ext_vector_type(2))) float v2f;
typedef __attribute__((ext_vector_type(8))) float v8f;

#define DDIM 2048
#define EEXP 64
#define TOPK 8
#define ROWS_PER_WG 64
#define WAVES_PER_WG 4
#define THREADS (WAVES_PER_WG * 32)

__global__ __launch_bounds__(THREADS)
void NoisyTopkRouter_28870770164343_kernel(
    const float* __restrict__ x,       // [M, D]
    const float* __restrict__ Wr,      // [E, D]
    const float* __restrict__ br,      // [E]
    const float* __restrict__ Wn,      // [E, D]
    const float* __restrict__ bn,      // [E]
    const float* __restrict__ noise,   // [M, E]
    float* __restrict__ out_weighted,  // [M, 8]
    float* __restrict__ out_indices,   // [M, 8] (float-encoded ints)
    float* __restrict__ out_softmax,   // [M, E]
    int Mtotal)
{
    // [mat][row][col] : mat 0 = route logits, mat 1 = noise logits
    __shared__ float lds[2 * ROWS_PER_WG * EEXP];   // 32 KB

    const int lane = threadIdx.x & 31;
    const int wave = threadIdx.x >> 5;
    const int wgRow0 = blockIdx.x * ROWS_PER_WG;
    const int waveRow0 = wgRow0 + wave * 16;

    // ---- WMMA f32 16x16x4 operand addressing (wave32 layout) ----
    // A (16x4): lane L<16 holds (M=L, K={k,k+1}); lane L>=16 holds (M=L-16, K={k+2,k+3})
    const int mrow  = lane & 15;
    const int khalf = (lane >> 4) * 2;
    const float* xa = x + (size_t)(waveRow0 + mrow) * DDIM + khalf;

    // B (4x16): lane n = L%16, same K split; element (k,n) = W[e0+n, k]
    const float* wb[2][4];
#pragma unroll
    for (int t = 0; t < 4; ++t) {
        const int e = t * 16 + mrow;
        wb[0][t] = Wr + (size_t)e * DDIM + khalf;
        wb[1][t] = Wn + (size_t)e * DDIM + khalf;
    }

    v8f acc[2][4];
#pragma unroll
    for (int m = 0; m < 2; ++m)
#pragma unroll
        for (int t = 0; t < 4; ++t)
            acc[m][t] = (v8f){0.f,0.f,0.f,0.f,0.f,0.f,0.f,0.f};

    // ---- K loop: 2048 in chunks of 16, 4 WMMA K-steps per chunk ----
    for (int k = 0; k < DDIM; k += 16) {
        __builtin_prefetch(xa + k + 128, 0, 1);   // stream x ahead (global_prefetch_b8)
#pragma unroll
        for (int kk = 0; kk < 16; kk += 4) {
            const v2f a = *(const v2f*)(xa + k + kk);
#pragma unroll
            for (int m = 0; m < 2; ++m) {
#pragma unroll
                for (int t = 0; t < 4; ++t) {
                    const v2f b = *(const v2f*)(wb[m][t] + k + kk);
                    acc[m][t] = __builtin_amdgcn_wmma_f32_16x16x4_f32(
                        false, a, false, b, (short)0, acc[m][t], false, false);
                }
            }
        }
    }

    // ---- Spill C tiles to LDS ----
    // C/D layout: lane L, VGPR v -> row M = v + 8*(L/16), col N = L%16 (+16*t)
#pragma unroll
    for (int m = 0; m < 2; ++m)
#pragma unroll
        for (int t = 0; t < 4; ++t)
#pragma unroll
            for (int v = 0; v < 8; ++v) {
                const int r = wave * 16 + v + 8 * (lane >> 4);
                const int c = (lane & 15) + 16 * t;
                lds[m * ROWS_PER_WG * EEXP + r * EEXP + c] = acc[m][t][v];
            }
    __syncthreads();

    // ---- Epilogue: one thread per row ----
    if (threadIdx.x < ROWS_PER_WG) {
        const int rl = threadIdx.x;
        const size_t row = (size_t)wgRow0 + rl;
        float* lg = &lds[rl * EEXP];
        const float* ns = &lds[ROWS_PER_WG * EEXP + rl * EEXP];
        const float* nz = noise + row * EEXP;

        // noisy logits = logit + noise * softplus(noise_logit); track max
        float mx = -INFINITY;
        for (int e = 0; e < EEXP; ++e) {
            const float l  = lg[e] + br[e];
            const float nl = ns[e] + bn[e];
            const float sp = fmaxf(nl, 0.0f) + log1pf(expf(-fabsf(nl)));
            const float v  = l + nz[e] * sp;
            lg[e] = v;
            mx = fmaxf(mx, v);
        }
        // softmax
        float sum = 0.0f;
        for (int e = 0; e < EEXP; ++e) {
            const float p = expf(lg[e] - mx);
            lg[e] = p;
            sum += p;
        }
        const float inv = 1.0f / sum;
        float* sm = out_softmax + row * EEXP;
        for (int e = 0; e < EEXP; ++e) {
            const float p = lg[e] * inv;
            lg[e] = p;
            sm[e] = p;
        }
        // top-8 (repeated argmax; strict > keeps lowest index on ties)
        float tv[TOPK];
        int   ti[TOPK];
        float tsum = 0.0f;
        for (int kk = 0; kk < TOPK; ++kk) {
            float best = -1.0f;
            int bi = 0;
            for (int e = 0; e < EEXP; ++e) {
                const float p = lg[e];
                if (p > best) { best = p; bi = e; }
            }
            tv[kk] = best;
            ti[kk] = bi;
            tsum += best;
            lg[bi] = -1.0f;
        }
        const float itw = 1.0f / tsum;
        float* ow = out_weighted + row * TOPK;
        float* oi = out_indices + row * TOPK;
        for (int kk = 0; kk < TOPK; ++kk) {
            ow[kk] = tv[kk] * itw;
            oi[kk] = (float)ti[kk];
        }
    }
}

extern "C" void kernel_launch(void* const* d_in, const int* in_sizes, int n_in,
                              void* d_out, int out_size, void* d_ws, size_t ws_size,
                              hipStream_t stream) {
    const float* x       = (const float*)d_in[0];
    const float* W_route = (const float*)d_in[1];
    const float* b_route = (const float*)d_in[2];
    const float* W_noise = (const float*)d_in[3];
    const float* b_noise = (const float*)d_in[4];
    const float* noise   = (const float*)d_in[5];

    const int Mtotal = in_sizes[0] / DDIM;   // B*S = 16384

    float* out = (float*)d_out;
    float* out_weighted = out;                         // M*8
    float* out_indices  = out + (size_t)Mtotal * TOPK; // M*8
    float* out_softmax  = out + (size_t)Mtotal * TOPK * 2; // M*64

    const int grid = Mtotal / ROWS_PER_WG;  // 256 workgroups
    NoisyTopkRouter_28870770164343_kernel<<<grid, THREADS, 0, stream>>>(
        x, W_route, b_route, W_noise, b_noise, noise,
        out_weighted, out_indices, out_softmax, Mtotal);
}